// HiddenClsAttentionBridge_37048387895868
// MI455X (gfx1250) — compile-verified
//
#include <hip/hip_runtime.h>
#include <math.h>

typedef __attribute__((ext_vector_type(2))) float v2f;
typedef __attribute__((ext_vector_type(8))) float v8f;

#define DIM 768
#define BD 16
#define SEQ 577
#define NPATCH 576
#define NB 64
#define HSTR 772   // padded LDS row stride (floats): 772 % 64 == 4 -> conflict-free A-frag reads

// ---------------------------------------------------------------------------
// Kernel 1: per-batch 16-row patch tile: LayerNorm + patch gate + K/V
// projection via fp32 WMMA (V_WMMA_F32_16X16X4_F32), K split over 8 waves.
// ---------------------------------------------------------------------------
__global__ __launch_bounds__(256) void kern_patch(
    const float* __restrict__ x, const float* __restrict__ ln_g, const float* __restrict__ ln_b,
    const float* __restrict__ Wk, const float* __restrict__ Wv,
    const float* __restrict__ pgate_w, const float* __restrict__ pgate_b,
    float* __restrict__ kout, float* __restrict__ vout, float* __restrict__ pgout)
{
    __shared__ float hid[16 * HSTR];
    __shared__ float accK[256];
    __shared__ float accV[256];
    const int b = blockIdx.y, tile = blockIdx.x;
    const int t = threadIdx.x;
    const int row = t >> 4, sub = t & 15;               // 16 threads per row
    const int srow = 1 + tile * 16 + row;               // global seq index (patches start at 1)
    const float* xr = x + ((size_t)b * SEQ + srow) * DIM;

    // pass 1: load row + moments
    float sum = 0.f, sumsq = 0.f;
    #pragma unroll 8
    for (int j = 0; j < 48; j++) {
        int d = sub + 16 * j;
        float v = xr[d];
        hid[row * HSTR + d] = v;
        sum += v; sumsq += v * v;
    }
    #pragma unroll
    for (int m = 8; m >= 1; m >>= 1) {
        sum   += __shfl_xor(sum, m, 32);
        sumsq += __shfl_xor(sumsq, m, 32);
    }
    const float mu   = sum * (1.f / 768.f);
    const float rstd = rsqrtf(sumsq * (1.f / 768.f) - mu * mu + 1e-5f);

    // pass 2: normalize in place + patch-gate dot
    float pgl = 0.f;
    #pragma unroll 8
    for (int j = 0; j < 48; j++) {
        int d = sub + 16 * j;
        float h = (hid[row * HSTR + d] - mu) * rstd * ln_g[d] + ln_b[d];
        hid[row * HSTR + d] = h;
        pgl += h * pgate_w[d];
    }
    #pragma unroll
    for (int m = 8; m >= 1; m >>= 1) pgl += __shfl_xor(pgl, m, 32);
    if (sub == 0)
        pgout[(size_t)b * NPATCH + tile * 16 + row] =
            1.f / (1.f + expf(-(pgl + pgate_b[0])));

    accK[t] = 0.f; accV[t] = 0.f;
    __syncthreads();

    // WMMA phase: wave w handles K-range [w*96, w*96+96) in 24 chunks of 4.
    // fp32 WMMA layouts (ISA 7.12.2): A 16x4: lanes0-15 hold K={kb,kb+1} (VGPR0/1),
    // lanes16-31 hold K={kb+2,kb+3}. B 4x16 symmetric. C/D: vgpr j -> M=j+8*half, N=r.
    const int lane = t & 31;
    const int half = lane >> 4, r = lane & 15;
    const int kb0 = (t >> 5) * 96;
    const float* hb  = &hid[r * HSTR];
    const float* wkb = Wk + r * DIM;
    const float* wvb = Wv + r * DIM;
    v8f ck = {0.f,0.f,0.f,0.f,0.f,0.f,0.f,0.f};
    v8f cv = {0.f,0.f,0.f,0.f,0.f,0.f,0.f,0.f};
    for (int i = 0; i < 24; i++) {
        int kk = kb0 + i * 4 + 2 * half;
        v2f a  = *reinterpret_cast<const v2f*>(hb  + kk);
        v2f bk = *reinterpret_cast<const v2f*>(wkb + kk);
        v2f bv = *reinterpret_cast<const v2f*>(wvb + kk);
        ck = __builtin_amdgcn_wmma_f32_16x16x4_f32(false, a, false, bk, (short)0, ck, false, false);
        cv = __builtin_amdgcn_wmma_f32_16x16x4_f32(false, a, false, bv, (short)0, cv, false, false);
    }
    #pragma unroll
    for (int j = 0; j < 8; j++) {
        atomicAdd(&accK[(j + 8 * half) * 16 + r], ck[j]);
        atomicAdd(&accV[(j + 8 * half) * 16 + r], cv[j]);
    }
    __syncthreads();

    const size_t base = ((size_t)b * NPATCH + tile * 16) * BD;
    kout[base + t] = accK[t];   // t = m*16 + n  ->  [row][bd]
    vout[base + t] = accV[t];
}

// ---------------------------------------------------------------------------
// Kernel 2: cls row: LayerNorm + q, c projections + gate logit
// ---------------------------------------------------------------------------
__global__ __launch_bounds__(256) void kern_cls(
    const float* __restrict__ x, const float* __restrict__ ln_g, const float* __restrict__ ln_b,
    const float* __restrict__ Wq, const float* __restrict__ Wc, const float* __restrict__ gate_w,
    float* __restrict__ qout, float* __restrict__ cout, float* __restrict__ glout)
{
    __shared__ float hid[DIM];
    __shared__ float red[256];
    const int b = blockIdx.x, t = threadIdx.x;
    const float* xr = x + (size_t)b * SEQ * DIM;
    float v0 = xr[t], v1 = xr[t + 256], v2 = xr[t + 512];

    red[t] = v0 + v1 + v2; __syncthreads();
    for (int st = 128; st; st >>= 1) { if (t < st) red[t] += red[t + st]; __syncthreads(); }
    const float mu = red[0] * (1.f / 768.f); __syncthreads();

    red[t] = v0*v0 + v1*v1 + v2*v2; __syncthreads();
    for (int st = 128; st; st >>= 1) { if (t < st) red[t] += red[t + st]; __syncthreads(); }
    const float rstd = rsqrtf(red[0] * (1.f / 768.f) - mu * mu + 1e-5f); __syncthreads();

    hid[t]       = (v0 - mu) * rstd * ln_g[t]       + ln_b[t];
    hid[t + 256] = (v1 - mu) * rstd * ln_g[t + 256] + ln_b[t + 256];
    hid[t + 512] = (v2 - mu) * rstd * ln_g[t + 512] + ln_b[t + 512];
    __syncthreads();

    const int g = t >> 4, sub = t & 15;    // group g computes output g
    float aq = 0.f, ac = 0.f, gl = 0.f;
    #pragma unroll 8
    for (int j = 0; j < 48; j++) {
        int d = sub + 16 * j;
        float h = hid[d];
        aq += h * Wq[g * DIM + d];
        ac += h * Wc[g * DIM + d];
        gl += h * gate_w[d];
    }
    #pragma unroll
    for (int m = 8; m >= 1; m >>= 1) {
        aq += __shfl_xor(aq, m, 32);
        ac += __shfl_xor(ac, m, 32);
        gl += __shfl_xor(gl, m, 32);
    }
    if (sub == 0) {
        qout[b * 16 + g] = aq;
        cout[b * 16 + g] = ac;
        if (g == 0) glout[b] = gl;
    }
}

// ---------------------------------------------------------------------------
// Kernel 3: attention (1 query x 576 keys), softmax, GELU summary, gates,
// and per-batch 768-d "ready" delta vectors (fold cls_gate/layer_scale/0.5).
// ---------------------------------------------------------------------------
__global__ __launch_bounds__(256) void kern_attn(
    const float* __restrict__ kbuf, const float* __restrict__ vbuf,
    const float* __restrict__ qbuf, const float* __restrict__ cbuf,
    const float* __restrict__ glbuf, const float* __restrict__ gate_b,
    const float* __restrict__ layer_scale,
    const float* __restrict__ Wo, const float* __restrict__ Wp,
    float* __restrict__ readyCls, float* __restrict__ readyPatch)
{
    __shared__ float qs[16], cs[16], summary[16];
    __shared__ float red[256];
    __shared__ float cred[256 * 16];
    __shared__ float sgate;
    const int b = blockIdx.x, t = threadIdx.x;
    if (t < 16) { qs[t] = qbuf[b * 16 + t]; cs[t] = cbuf[b * 16 + t]; }
    __syncthreads();

    const float* kb = kbuf + (size_t)b * NPATCH * BD;
    const float* vb = vbuf + (size_t)b * NPATCH * BD;
    float sc[3];
    float mymax = -3.4e38f;
    #pragma unroll
    for (int i = 0; i < 3; i++) {
        int n = t + 256 * i;
        float s = -3.4e38f;
        if (n < NPATCH) {
            s = 0.f;
            #pragma unroll
            for (int k2 = 0; k2 < 16; k2++) s += qs[k2] * kb[n * 16 + k2];
            s *= 0.25f;   // SCALE = BD^-0.5
        }
        sc[i] = s;
        mymax = fmaxf(mymax, s);
    }
    red[t] = mymax; __syncthreads();
    for (int st = 128; st; st >>= 1) { if (t < st) red[t] = fmaxf(red[t], red[t + st]); __syncthreads(); }
    const float M = red[0]; __syncthreads();

    float ctx[16];
    #pragma unroll
    for (int k2 = 0; k2 < 16; k2++) ctx[k2] = 0.f;
    float psum = 0.f;
    #pragma unroll
    for (int i = 0; i < 3; i++) {
        int n = t + 256 * i;
        if (n < NPATCH) {
            float p = expf(sc[i] - M);
            psum += p;
            #pragma unroll
            for (int k2 = 0; k2 < 16; k2++) ctx[k2] += p * vb[n * 16 + k2];
        }
    }
    red[t] = psum; __syncthreads();
    for (int st = 128; st; st >>= 1) { if (t < st) red[t] += red[t + st]; __syncthreads(); }
    const float Z = red[0]; __syncthreads();

    #pragma unroll
    for (int k2 = 0; k2 < 16; k2++) cred[t * 16 + k2] = ctx[k2];
    __syncthreads();
    for (int st = 128; st; st >>= 1) {
        if (t < st) {
            #pragma unroll
            for (int k2 = 0; k2 < 16; k2++) cred[t * 16 + k2] += cred[(t + st) * 16 + k2];
        }
        __syncthreads();
    }
    if (t < 16) {
        float c = cred[t] / Z + cs[t];
        summary[t] = 0.5f * c * (1.f + erff(c * 0.70710678118654752f));  // exact GELU
    }
    if (t == 0) sgate = 1.f / (1.f + expf(-(glbuf[b] + gate_b[0])));
    __syncthreads();

    const float g = sgate;
    #pragma unroll
    for (int i = 0; i < 3; i++) {
        int d = t + 256 * i;
        float so = 0.f, sp = 0.f;
        #pragma unroll
        for (int k2 = 0; k2 < 16; k2++) {
            float s = summary[k2];
            so += s * Wo[d * 16 + k2];
            sp += s * Wp[d * 16 + k2];
        }
        readyCls[b * DIM + d]   = g * layer_scale[d] * so;      // RESIDUAL_SCALE = 1
        readyPatch[b * DIM + d] = 0.5f * g * sp;                // PATCH_FEEDBACK_SCALE = 0.5
    }
}

// ---------------------------------------------------------------------------
// Kernel 4: streaming residual add (bandwidth-bound pass, float4).
// ---------------------------------------------------------------------------
__global__ __launch_bounds__(256) void kern_out(
    const float4* __restrict__ x4, const float4* __restrict__ rc4,
    const float4* __restrict__ rp4, const float* __restrict__ pg,
    float4* __restrict__ out4)
{
    const int i = blockIdx.x * 256 + threadIdx.x;
    const int total = NB * SEQ * (DIM / 4);
    if (i >= total) return;
    const int perB = SEQ * (DIM / 4);
    const int b   = i / perB;
    const int rem = i - b * perB;
    const int s   = rem / (DIM / 4);
    const int dq  = rem - s * (DIM / 4);
    float4 xv = x4[i];
    if (s == 0) {
        float4 dv = rc4[b * (DIM / 4) + dq];
        xv.x += dv.x; xv.y += dv.y; xv.z += dv.z; xv.w += dv.w;
    } else {
        float gm = pg[b * NPATCH + (s - 1)];
        float4 dv = rp4[b * (DIM / 4) + dq];
        xv.x += gm * dv.x; xv.y += gm * dv.y; xv.z += gm * dv.z; xv.w += gm * dv.w;
    }
    out4[i] = xv;
}

// ---------------------------------------------------------------------------
extern "C" void kernel_launch(void* const* d_in, const int* in_sizes, int n_in,
                              void* d_out, int out_size, void* d_ws, size_t ws_size,
                              hipStream_t stream)
{
    const float* x           = (const float*)d_in[0];
    const float* ln_g        = (const float*)d_in[1];
    const float* ln_b        = (const float*)d_in[2];
    const float* Wq          = (const float*)d_in[3];
    const float* Wk          = (const float*)d_in[4];
    const float* Wv          = (const float*)d_in[5];
    const float* Wc          = (const float*)d_in[6];
    const float* Wo          = (const float*)d_in[7];
    const float* Wp          = (const float*)d_in[8];
    const float* gate_w      = (const float*)d_in[9];
    const float* gate_b      = (const float*)d_in[10];
    const float* pgate_w     = (const float*)d_in[11];
    const float* pgate_b     = (const float*)d_in[12];
    const float* layer_scale = (const float*)d_in[13];
    float* out = (float*)d_out;

    float* ws   = (float*)d_ws;
    float* wsK  = ws;                          // 64*576*16 = 589824
    float* wsV  = wsK + 589824;                // 589824
    float* wsQ  = wsV + 589824;                // 1024
    float* wsC  = wsQ + 1024;                  // 1024
    float* wsGL = wsC + 1024;                  // 64
    float* wsPG = wsGL + 64;                   // 36864
    float* wsRC = wsPG + 36864;                // 49152 (16B-aligned offset)
    float* wsRP = wsRC + 49152;                // 49152
    // total ~5.3 MB of workspace

    dim3 gA(36, NB);
    kern_patch<<<gA, 256, 0, stream>>>(x, ln_g, ln_b, Wk, Wv, pgate_w, pgate_b,
                                       wsK, wsV, wsPG);
    kern_cls<<<NB, 256, 0, stream>>>(x, ln_g, ln_b, Wq, Wc, gate_w, wsQ, wsC, wsGL);
    kern_attn<<<NB, 256, 0, stream>>>(wsK, wsV, wsQ, wsC, wsGL, gate_b, layer_scale,
                                      Wo, Wp, wsRC, wsRP);
    const int total4 = NB * SEQ * (DIM / 4);
    kern_out<<<(total4 + 255) / 256, 256, 0, stream>>>(
        (const float4*)x, (const float4*)wsRC, (const float4*)wsRP, wsPG, (float4*)out);
}